// ginNet_59098749993119
// MI455X (gfx1250) — compile-verified
//
#include <hip/hip_runtime.h>
#include <hip/hip_bf16.h>
#include <math.h>

typedef float v2f __attribute__((ext_vector_type(2)));
typedef float v8f __attribute__((ext_vector_type(8)));

#define BN_EPS 1e-5f

// ---------------------------------------------------------------- zero
__global__ void zero_f32(float* __restrict__ p, long n) {
  long i = (long)blockIdx.x * blockDim.x + threadIdx.x;
  long stride = (long)gridDim.x * blockDim.x;
  for (; i < n; i += stride) p[i] = 0.0f;
}

// ------------------------------------------------------------- init agg
// agg[n*astride + k] = (k < bvalid) ? base[n*bstride + k] : 0   (astride = 1<<kshift)
__global__ void init_agg(const float* __restrict__ base, int bstride, int bvalid,
                         float* __restrict__ agg, int kshift, long total) {
  long i = (long)blockIdx.x * blockDim.x + threadIdx.x;
  long stride = (long)gridDim.x * blockDim.x;
  int kmask = (1 << kshift) - 1;
  for (; i < total; i += stride) {
    long n = i >> kshift;
    int k = (int)(i & kmask);
    agg[i] = (k < bvalid) ? base[n * (long)bstride + k] : 0.0f;
  }
}

// ---------------------------------------------------------- scatter-add
// grid.x covers edges, grid.y = feature chunk (4 features per chunk).
__global__ void scatter_add(const long long* __restrict__ src,
                            const long long* __restrict__ dst,
                            const float* __restrict__ xin, int in_stride,
                            float* __restrict__ agg, int agg_stride,
                            int din, long E) {
  long e = (long)blockIdx.x * blockDim.x + threadIdx.x;
  if (e >= E) return;
  int f0 = blockIdx.y * 4;
  long s = src[e];
  long d = dst[e];
  const float* xr = xin + s * (long)in_stride;
  float* ar = agg + d * (long)agg_stride;
  if (((in_stride & 3) == 0) && (f0 + 4 <= din)) {
    float4 v = *(const float4*)(xr + f0);      // aligned: stride mult of 4, f0 mult of 4
    atomicAdd(&ar[f0 + 0], v.x);
    atomicAdd(&ar[f0 + 1], v.y);
    atomicAdd(&ar[f0 + 2], v.z);
    atomicAdd(&ar[f0 + 3], v.w);
  } else {
    int fend = min(f0 + 4, din);
    for (int f = f0; f < fend; ++f)
      atomicAdd(&ar[f], xr[f]);
  }
}

// -------------------------------------------------- fused GIN MLP (WMMA)
// hout = relu( relu(hin @ W1 + b1) @ W2 + b2 ), plus BN stat sums.
// hin has row stride DINPAD (already = x + scatter-sum, zero padded).
// 256 threads = 8 waves; each wave owns a 16-node row tile.
template <int DINPAD>
__global__ __launch_bounds__(256)
void gin_mlp(const float* __restrict__ hin,
             const float* __restrict__ W1g, int k1,   // k1 = din (<= DINPAD)
             const float* __restrict__ b1g,
             const float* __restrict__ W2g,
             const float* __restrict__ b2g,
             float* __restrict__ hout,
             float* __restrict__ stats,   // [64]: sum[0..31], sumsq[32..63]
             int N) {
  constexpr int PITCH = DINPAD + 1;
  __shared__ float sW1[64 * 32];
  __shared__ float sB1[32];
  __shared__ float sW2[32 * 32];
  __shared__ float sB2[32];
  __shared__ float sStage[8][16 * PITCH];
  __shared__ float sSum[32];
  __shared__ float sSq[32];

  const int tid = threadIdx.x;
  for (int i = tid; i < DINPAD * 32; i += 256) {
    int k = i >> 5, n = i & 31;
    sW1[i] = (k < k1) ? W1g[k * 32 + n] : 0.0f;
  }
  for (int i = tid; i < 32 * 32; i += 256) sW2[i] = W2g[i];
  if (tid < 32) { sB1[tid] = b1g[tid]; sB2[tid] = b2g[tid]; sSum[tid] = 0.0f; sSq[tid] = 0.0f; }
  __syncthreads();

  const int wid  = tid >> 5;          // wave id (wave32)
  const int lane = tid & 31;
  const int l16  = lane & 15;
  const int hi   = lane >> 4;         // 0: K pair (k,k+1)  1: K pair (k+2,k+3)

  const long node_base = ((long)blockIdx.x * 8 + wid) * 16;
  float* st = sStage[wid];

  // ---- stage 16 x DINPAD input tile into wave-private LDS (float4 loads)
  {
    constexpr int CH = DINPAD / 4;           // 4-float chunks per row
#pragma unroll
    for (int c = lane; c < 16 * CH; c += 32) {
      int r  = c / CH;
      int k4 = (c - r * CH) * 4;
      long rr = node_base + r;
      float4 v = make_float4(0.f, 0.f, 0.f, 0.f);
      if (rr < N) v = *(const float4*)(hin + rr * DINPAD + k4);
      st[r * PITCH + k4 + 0] = v.x;
      st[r * PITCH + k4 + 1] = v.y;
      st[r * PITCH + k4 + 2] = v.z;
      st[r * PITCH + k4 + 3] = v.w;
    }
  }
  // intra-wave DS ordering: stores above complete before loads below (same wave, in-order DS)

  v8f c0 = {0.f,0.f,0.f,0.f,0.f,0.f,0.f,0.f};
  v8f c1 = c0;
  // ---- GEMM1: in[16 x DINPAD] @ W1[DINPAD x 32]  (fully unrolled)
#pragma unroll
  for (int k0 = 0; k0 < DINPAD; k0 += 4) {
    int kA = k0 + hi * 2;
    v2f a   = { st[l16 * PITCH + kA],       st[l16 * PITCH + kA + 1] };
    v2f bN0 = { sW1[kA * 32 + l16],         sW1[(kA + 1) * 32 + l16] };
    v2f bN1 = { sW1[kA * 32 + 16 + l16],    sW1[(kA + 1) * 32 + 16 + l16] };
    c0 = __builtin_amdgcn_wmma_f32_16x16x4_f32(false, a, false, bN0, (short)0, c0, false, false);
    c1 = __builtin_amdgcn_wmma_f32_16x16x4_f32(false, a, false, bN1, (short)0, c1, false, false);
  }
  // bias + relu; columns fixed per lane in C layout
  {
    float bc0 = sB1[l16], bc1 = sB1[16 + l16];
#pragma unroll
    for (int r = 0; r < 8; ++r) {
      c0[r] = fmaxf(c0[r] + bc0, 0.0f);
      c1[r] = fmaxf(c1[r] + bc1, 0.0f);
    }
  }
  // stage h1 (C layout -> row-major) back into the same wave-private slab
#pragma unroll
  for (int r = 0; r < 8; ++r) {
    int M = r + hi * 8;                 // C layout: lanes>=16 hold rows M=r+8
    st[M * PITCH + l16]      = c0[r];
    st[M * PITCH + 16 + l16] = c1[r];
  }

  // ---- GEMM2: h1[16 x 32] @ W2[32 x 32]  (fully unrolled)
  v8f d0 = {0.f,0.f,0.f,0.f,0.f,0.f,0.f,0.f};
  v8f d1 = d0;
#pragma unroll
  for (int k0 = 0; k0 < 32; k0 += 4) {
    int kA = k0 + hi * 2;
    v2f a   = { st[l16 * PITCH + kA],       st[l16 * PITCH + kA + 1] };
    v2f bN0 = { sW2[kA * 32 + l16],         sW2[(kA + 1) * 32 + l16] };
    v2f bN1 = { sW2[kA * 32 + 16 + l16],    sW2[(kA + 1) * 32 + 16 + l16] };
    d0 = __builtin_amdgcn_wmma_f32_16x16x4_f32(false, a, false, bN0, (short)0, d0, false, false);
    d1 = __builtin_amdgcn_wmma_f32_16x16x4_f32(false, a, false, bN1, (short)0, d1, false, false);
  }
  // bias + post-conv relu, store, accumulate BN stats
  {
    float bc0 = sB2[l16], bc1 = sB2[16 + l16];
    float s0 = 0.f, s1 = 0.f, q0 = 0.f, q1 = 0.f;
#pragma unroll
    for (int r = 0; r < 8; ++r) {
      int M = r + hi * 8;
      long orow = node_base + M;
      float v0 = fmaxf(d0[r] + bc0, 0.0f);
      float v1 = fmaxf(d1[r] + bc1, 0.0f);
      if (orow < N) {
        hout[orow * 32 + l16]      = v0;
        hout[orow * 32 + 16 + l16] = v1;
        s0 += v0; q0 += v0 * v0;
        s1 += v1; q1 += v1 * v1;
      }
    }
    atomicAdd(&sSum[l16],      s0);
    atomicAdd(&sSum[16 + l16], s1);
    atomicAdd(&sSq[l16],       q0);
    atomicAdd(&sSq[16 + l16],  q1);
  }
  __syncthreads();
  if (tid < 32) {
    atomicAdd(&stats[tid],      sSum[tid]);
    atomicAdd(&stats[32 + tid], sSq[tid]);
  }
}

// ------------------------------------------------------------ batchnorm
__global__ void bn_apply(float* __restrict__ h, const float* __restrict__ stats,
                         const float* __restrict__ g, const float* __restrict__ b,
                         int N) {
  long total = (long)N * 32;
  long i = (long)blockIdx.x * blockDim.x + threadIdx.x;
  long stride = (long)gridDim.x * blockDim.x;
  float invN = 1.0f / (float)N;
  for (; i < total; i += stride) {
    int c = (int)(i & 31);
    float mean = stats[c] * invN;
    float var  = stats[32 + c] * invN - mean * mean;
    h[i] = (h[i] - mean) * rsqrtf(var + BN_EPS) * g[c] + b[c];
  }
}

// ------------------------------------------------------------------ head
__global__ __launch_bounds__(256)
void head(const float* __restrict__ h,
          const float* __restrict__ fc1W, const float* __restrict__ fc1b,
          const float* __restrict__ fc2W, const float* __restrict__ fc2b,
          float* __restrict__ out, int N) {
  __shared__ float sW1[32 * 32];
  __shared__ float sB1[32];
  __shared__ float sW2[32 * 2];
  __shared__ float sB2[2];
  for (int i = threadIdx.x; i < 32 * 32; i += 256) sW1[i] = fc1W[i];
  if (threadIdx.x < 32) sB1[threadIdx.x] = fc1b[threadIdx.x];
  if (threadIdx.x < 64) sW2[threadIdx.x] = fc2W[threadIdx.x];
  if (threadIdx.x < 2)  sB2[threadIdx.x] = fc2b[threadIdx.x];
  __syncthreads();
  long n = (long)blockIdx.x * blockDim.x + threadIdx.x;
  if (n >= N) return;
  float hv[32];
#pragma unroll
  for (int i = 0; i < 32; ++i) hv[i] = h[n * 32 + i];
  float l0 = sB2[0], l1 = sB2[1];
  for (int j = 0; j < 32; ++j) {
    float t = sB1[j];
#pragma unroll
    for (int i = 0; i < 32; ++i) t += hv[i] * sW1[i * 32 + j];
    t = fmaxf(t, 0.0f);
    l0 += t * sW2[j * 2 + 0];
    l1 += t * sW2[j * 2 + 1];
  }
  float m = fmaxf(l0, l1);
  float lse = m + logf(expf(l0 - m) + expf(l1 - m));
  out[n * 2 + 0] = l0 - lse;
  out[n * 2 + 1] = l1 - lse;
}

// ---------------------------------------------------------------- launch
extern "C" void kernel_launch(void* const* d_in, const int* in_sizes, int n_in,
                              void* d_out, int out_size, void* d_ws, size_t ws_size,
                              hipStream_t stream) {
  const float* x = (const float*)d_in[0];
  const long long* ei = (const long long*)d_in[1];   // int64 edge_index (2,E)
  const int N = in_sizes[0] / 61;
  const long E = (long)in_sizes[1] / 2;
  const long long* src = ei;        // edge_index[0]
  const long long* dst = ei + E;    // edge_index[1]

  auto P = [&](int i) { return (const float*)d_in[2 + i]; };
  // dict order: for l in 0..4: W1,b1,W2,b2,bn_g,bn_b ; then fc1_W,fc1_b,fc2_W,fc2_b

  float* agg   = (float*)d_ws;                       // N*64
  float* hA    = agg + (size_t)N * 64;               // N*32
  float* hB    = hA + (size_t)N * 32;                // N*32
  float* stats = hB + (size_t)N * 32;                // 64 floats

  const float* hin = x;
  int bstride = 61, bvalid = 61;
  float* nxt = hA;

  for (int l = 0; l < 5; ++l) {
    const int din = (l == 0) ? 61 : 32;
    const int kshift = (l == 0) ? 6 : 5;
    const int astride = 1 << kshift;

    long zn = (long)N * astride;
    init_agg<<<(unsigned)((zn + 255) / 256), 256, 0, stream>>>(
        hin, bstride, bvalid, agg, kshift, zn);
    zero_f32<<<1, 64, 0, stream>>>(stats, 64);

    dim3 sgrid((unsigned)((E + 255) / 256), (unsigned)((din + 3) / 4));
    scatter_add<<<sgrid, 256, 0, stream>>>(src, dst, hin, bstride, agg, astride, din, E);

    if (l == 0) {
      gin_mlp<64><<<(unsigned)((N + 127) / 128), 256, 0, stream>>>(
          agg, P(0), 61, P(1), P(2), P(3), nxt, stats, N);
    } else {
      gin_mlp<32><<<(unsigned)((N + 127) / 128), 256, 0, stream>>>(
          agg, P(l * 6 + 0), 32, P(l * 6 + 1), P(l * 6 + 2), P(l * 6 + 3), nxt, stats, N);
    }

    bn_apply<<<(unsigned)(((long)N * 32 + 255) / 256), 256, 0, stream>>>(
        nxt, stats, P(l * 6 + 4), P(l * 6 + 5), N);

    hin = nxt; bstride = 32; bvalid = 32;
    nxt = (nxt == hA) ? hB : hA;
  }

  head<<<(unsigned)((N + 255) / 256), 256, 0, stream>>>(
      hin, P(30), P(31), P(32), P(33), (float*)d_out, N);
}